// DotProductAttention_54125177865167
// MI455X (gfx1250) — compile-verified
//
#include <hip/hip_runtime.h>
#include <hip/hip_bf16.h>

typedef float v2f __attribute__((ext_vector_type(2)));
typedef float v8f __attribute__((ext_vector_type(8)));

#define B_   4
#define S_   2048
#define DM_  1024
#define DK_  128
#define KC_  64                    // K-chunk staged in LDS
#define LDP2_ (2 * DK_ + 32)       // pair-interleaved LDS row stride (floats)
#define NEG_INF (-__builtin_inff())

__device__ __forceinline__ v8f wmma4(v2f a, v2f b, v8f c) {
    return __builtin_amdgcn_wmma_f32_16x16x4_f32(false, a, false, b,
                                                 (short)0, c, false, false);
}

// Stage rows [kc, kc+KC_) of a [*,128] row-major matrix into LDS,
// k-pair interleaved: Wl[p*LDP2_ + col*2 + (k&1)] = W[kc+2p+(k&1)][col].
// 256 threads, 4 iters, float4 in / 2x float4 out, all 16B aligned.
__device__ __forceinline__ void stage_pairs(float* __restrict__ Wl,
                                            const float* __restrict__ W,
                                            int kc, int tid) {
#pragma unroll
    for (int i = 0; i < 4; ++i) {
        const int idx = i * 256 + tid;     // 0..1023
        const int p   = idx >> 5;          // pair-row 0..31
        const int c4  = idx & 31;          // float4 column group
        float4 r0 = *(const float4*)(W + (size_t)(kc + 2 * p)     * DK_ + c4 * 4);
        float4 r1 = *(const float4*)(W + (size_t)(kc + 2 * p + 1) * DK_ + c4 * 4);
        float* d = Wl + p * LDP2_ + c4 * 8;
        *(float4*)(d)     = make_float4(r0.x, r1.x, r0.y, r1.y);
        *(float4*)(d + 4) = make_float4(r0.z, r1.z, r0.w, r1.w);
    }
}

// ---------------------------------------------------------------------------
// Kernel 1: Q/K/V projections.  X:[8192,1024] x W:[1024,128] -> [8192,128].
// 8 waves/block; each wave owns a 16-row strip and computes all 128 output
// columns (8 accumulators).  W staged in LDS (pair-interleaved) once per
// workgroup per 64-row K-chunk; B fragments are single ds_load_b64.
// grid = (8192/128, 3 weights), block = 256
// ---------------------------------------------------------------------------
__global__ void __launch_bounds__(256)
proj_kernel(const float* __restrict__ X,
            const float* __restrict__ Wq,
            const float* __restrict__ Wk,
            const float* __restrict__ Wv,
            float* __restrict__ Q,
            float* __restrict__ K,
            float* __restrict__ V) {
    __shared__ float Wl[(KC_ / 2) * LDP2_];       // 36 KB

    const int tid    = threadIdx.x;
    const int wave   = tid >> 5;
    const int lane   = tid & 31;
    const int laneLo = lane & 15;
    const int laneHi = lane >> 4;

    const int row0 = blockIdx.x * 128 + wave * 16;
    const int wsel = blockIdx.y;
    const float* __restrict__ W = (wsel == 0) ? Wq : (wsel == 1) ? Wk : Wv;
    float* __restrict__ Out     = (wsel == 0) ? Q  : (wsel == 1) ? K  : V;

    const float* arow = X + (size_t)(row0 + laneLo) * DM_ + 2 * laneHi;

    v8f c[8] = {};
    for (int kc = 0; kc < DM_; kc += KC_) {
        stage_pairs(Wl, W, kc, tid);
        __syncthreads();

#pragma unroll 2
        for (int kk = 0; kk < KC_; kk += 4) {
            v2f a = *(const v2f*)(arow + kc + kk);
            const float* brow = Wl + ((kk >> 1) + laneHi) * LDP2_ + laneLo * 2;
#pragma unroll
            for (int t = 0; t < 8; ++t) {
                v2f b = *(const v2f*)(brow + t * 32);   // {W[k][col], W[k+1][col]}
                c[t] = wmma4(a, b, c[t]);
            }
        }
        __syncthreads();
    }

    float* obase = Out + (size_t)(row0 + 8 * laneHi) * DK_ + laneLo;
#pragma unroll
    for (int t = 0; t < 8; ++t)
#pragma unroll
        for (int m = 0; m < 8; ++m)
            obase[(size_t)m * DK_ + t * 16] = c[t][m];
}

// ---------------------------------------------------------------------------
// Kernel 2: scores = Q K^T / sqrt(128) with causal mask -> attention region.
// One wave -> 32x32 output (2x2 tiles, 4 accums); A/B fragments are single
// contiguous b64 global loads, each feeding 2 WMMAs.
// grid = (S/32, S/32, B), block = 32.  Fully-masked blocks skip WMMA.
// ---------------------------------------------------------------------------
__global__ void __launch_bounds__(32)
scores_kernel(const float* __restrict__ Q,
              const float* __restrict__ Km,
              float* __restrict__ Attn) {
    const int lane   = threadIdx.x;
    const int laneLo = lane & 15;
    const int laneHi = lane >> 4;

    const int s0 = blockIdx.x * 32;
    const int q0 = blockIdx.y * 32;
    const int b  = blockIdx.z;

    float* base = Attn + (size_t)b * S_ * S_;

    if (s0 > q0 + 31) {                        // whole 32x32 block masked
#pragma unroll
        for (int qi = 0; qi < 2; ++qi)
#pragma unroll
            for (int si = 0; si < 2; ++si) {
                float* t = base + (size_t)(q0 + 16 * qi + 8 * laneHi) * S_
                                + s0 + 16 * si + laneLo;
#pragma unroll
                for (int m = 0; m < 8; ++m)
                    t[(size_t)m * S_] = NEG_INF;
            }
        return;
    }

    const float* qrow0 = Q  + (size_t)b * S_ * DK_ + (size_t)(q0 + laneLo) * DK_ + 2 * laneHi;
    const float* qrow1 = qrow0 + (size_t)16 * DK_;
    const float* krow0 = Km + (size_t)b * S_ * DK_ + (size_t)(s0 + laneLo) * DK_ + 2 * laneHi;
    const float* krow1 = krow0 + (size_t)16 * DK_;

    v8f c00 = {}, c01 = {}, c10 = {}, c11 = {};
#pragma unroll 8
    for (int k = 0; k < DK_; k += 4) {
        v2f a0 = *(const v2f*)(qrow0 + k);
        v2f a1 = *(const v2f*)(qrow1 + k);
        v2f b0 = *(const v2f*)(krow0 + k);     // B[k][n] = K[s][k]: contiguous
        v2f b1 = *(const v2f*)(krow1 + k);
        c00 = wmma4(a0, b0, c00);
        c01 = wmma4(a0, b1, c01);
        c10 = wmma4(a1, b0, c10);
        c11 = wmma4(a1, b1, c11);
    }

    const float scale = 0.08838834764831845f;  // 1/sqrt(128)
#pragma unroll
    for (int qi = 0; qi < 2; ++qi)
#pragma unroll
        for (int si = 0; si < 2; ++si) {
            const v8f& c = qi == 0 ? (si == 0 ? c00 : c01)
                                   : (si == 0 ? c10 : c11);
            const int s = s0 + 16 * si + laneLo;
            float* t = base + (size_t)(q0 + 16 * qi + 8 * laneHi) * S_
                            + s0 + 16 * si + laneLo;
#pragma unroll
            for (int m = 0; m < 8; ++m) {
                const int q = q0 + 16 * qi + m + 8 * laneHi;
                t[(size_t)m * S_] = (s <= q) ? c[m] * scale : NEG_INF;
            }
        }
}

// ---------------------------------------------------------------------------
// Kernel 3: per-column (query-axis) softmax stats: max, then 1/sum(exp).
// thread = key column s -> coalesced sweeps; prefetch streams the columns.
// grid = (S/256, B), block = 256
// ---------------------------------------------------------------------------
__global__ void __launch_bounds__(256)
colstats_kernel(const float* __restrict__ Attn,
                float* __restrict__ colmax,
                float* __restrict__ colsum) {
    const int s = blockIdx.x * 256 + threadIdx.x;
    const int b = blockIdx.y;
    const float* col = Attn + (size_t)b * S_ * S_ + s;

    float m = NEG_INF;
#pragma unroll 4
    for (int q = 0; q < S_; ++q) {
        if ((q & 15) == 0 && q + 64 < S_)
            __builtin_prefetch(col + (size_t)(q + 64) * S_, 0, 0);
        m = fmaxf(m, col[(size_t)q * S_]);
    }

    float sum = 0.0f;
#pragma unroll 4
    for (int q = 0; q < S_; ++q)
        sum += __expf(col[(size_t)q * S_] - m);    // exp(-inf - m) == 0

    colmax[b * S_ + s] = m;
    colsum[b * S_ + s] = 1.0f / sum;
}

// ---------------------------------------------------------------------------
// Kernel 4: in-place normalize: attn = exp(score - max_col) * rsum_col.
// float4 vectorized; each float4 shares (b,q), spans s..s+3.
// ---------------------------------------------------------------------------
__global__ void __launch_bounds__(256)
colnorm_kernel(float* __restrict__ Attn,
               const float* __restrict__ colmax,
               const float* __restrict__ colsum) {
    const size_t idx = (size_t)blockIdx.x * 256 + threadIdx.x;   // float4 idx
    const size_t lin = idx * 4;
    const int s0 = (int)(lin & (S_ - 1));
    const int b  = (int)(lin >> 22);                             // S_*S_ = 2^22
    const int sb = b * S_ + s0;

    float4 x  = *(float4*)(Attn + lin);
    float4 mx = *(const float4*)(colmax + sb);
    float4 rs = *(const float4*)(colsum + sb);
    x.x = __expf(x.x - mx.x) * rs.x;
    x.y = __expf(x.y - mx.y) * rs.y;
    x.z = __expf(x.z - mx.z) * rs.z;
    x.w = __expf(x.w - mx.w) * rs.w;
    *(float4*)(Attn + lin) = x;
}

// ---------------------------------------------------------------------------
// Kernel 5: out = attention @ V.  [2048,2048] x [2048,128] per batch.
// Same structure as proj: wave owns 16 q-rows x all 128 v-cols; V chunks
// staged pair-interleaved in LDS; attention rows read once.
// grid = (S/128, B), block = 256
// ---------------------------------------------------------------------------
__global__ void __launch_bounds__(256)
outgemm_kernel(const float* __restrict__ Attn,
               const float* __restrict__ V,
               float* __restrict__ Out) {
    __shared__ float Vl[(KC_ / 2) * LDP2_];       // 36 KB

    const int tid    = threadIdx.x;
    const int wave   = tid >> 5;
    const int lane   = tid & 31;
    const int laneLo = lane & 15;
    const int laneHi = lane >> 4;

    const int q0 = blockIdx.x * 128 + wave * 16;
    const int b  = blockIdx.y;

    const float* arow = Attn + (size_t)b * S_ * S_ + (size_t)(q0 + laneLo) * S_ + 2 * laneHi;
    const float* Vb   = V + (size_t)b * S_ * DK_;

    v8f c[8] = {};
    for (int kc = 0; kc < S_; kc += KC_) {
        stage_pairs(Vl, Vb, kc, tid);
        __syncthreads();

#pragma unroll 2
        for (int kk = 0; kk < KC_; kk += 4) {
            v2f a = *(const v2f*)(arow + kc + kk);
            const float* brow = Vl + ((kk >> 1) + laneHi) * LDP2_ + laneLo * 2;
#pragma unroll
            for (int t = 0; t < 8; ++t) {
                v2f bb = *(const v2f*)(brow + t * 32);
                c[t] = wmma4(a, bb, c[t]);
            }
        }
        __syncthreads();
    }

    float* obase = Out + (size_t)b * S_ * DK_ + (size_t)(q0 + 8 * laneHi) * DK_ + laneLo;
#pragma unroll
    for (int t = 0; t < 8; ++t)
#pragma unroll
        for (int m = 0; m < 8; ++m)
            obase[(size_t)m * DK_ + t * 16] = c[t][m];
}

// ---------------------------------------------------------------------------
extern "C" void kernel_launch(void* const* d_in, const int* in_sizes, int n_in,
                              void* d_out, int out_size, void* d_ws, size_t ws_size,
                              hipStream_t stream) {
    const float* X  = (const float*)d_in[0];   // [4,2048,1024]
    const float* Wq = (const float*)d_in[1];   // [1024,128]
    const float* Wk = (const float*)d_in[2];
    const float* Wv = (const float*)d_in[3];

    float* out  = (float*)d_out;                         // [4,2048,128]
    float* attn = (float*)d_out + (size_t)B_ * S_ * DK_; // [4,2048,2048]

    char* ws = (char*)d_ws;
    float* Q      = (float*)(ws);                                   // 4 MB
    float* K      = Q + (size_t)B_ * S_ * DK_;                      // 4 MB
    float* V      = K + (size_t)B_ * S_ * DK_;                      // 4 MB
    float* colmax = V + (size_t)B_ * S_ * DK_;                      // 32 KB
    float* colsum = colmax + (size_t)B_ * S_;                       // 32 KB

    // 1. projections (X read once per weight)
    proj_kernel<<<dim3((B_ * S_) / 128, 3), 256, 0, stream>>>(X, Wq, Wk, Wv, Q, K, V);
    // 2. masked scaled scores -> attention buffer
    scores_kernel<<<dim3(S_ / 32, S_ / 32, B_), 32, 0, stream>>>(Q, K, attn);
    // 3. column (query-axis) softmax stats
    colstats_kernel<<<dim3(S_ / 256, B_), 256, 0, stream>>>(attn, colmax, colsum);
    // 4. normalize in place
    {
        size_t total4 = (size_t)B_ * S_ * S_ / 4;
        colnorm_kernel<<<dim3((unsigned)(total4 / 256)), 256, 0, stream>>>(attn, colmax, colsum);
    }
    // 5. out = attention @ V
    outgemm_kernel<<<dim3(S_ / 128, B_), 256, 0, stream>>>(attn, V, out);
}